// NeuralNet_66666482368821
// MI455X (gfx1250) — compile-verified
//
#include <hip/hip_runtime.h>
#include <hip/hip_bf16.h>
#include <stdint.h>

#ifndef __has_builtin
#define __has_builtin(x) 0
#endif

#if __has_builtin(__builtin_amdgcn_tensor_load_to_lds) && __has_builtin(__builtin_amdgcn_s_wait_tensorcnt)
#define USE_TDM 1
#else
#define USE_TDM 0
#endif

#define B_DIM 8192
#define F_DIM 4096
#define TILE 64
#define LDS_STRIDE 65   // 64 + 1 pad dword per row -> bank-conflict-free transposed reads

// Piecewise cubic matching jnp.searchsorted(side='left') bucketing:
// interval promotes on strict v > breakpoint; v < -2 -> 0.
__device__ __forceinline__ float piecewise(float v) {
    float c0 = 0.5f, c1 = -1.0f, c2 = 0.25f, c3 = 0.10f;
    if (v > -1.0f) { c0 = 0.0f;  c1 = 1.0f;  c2 = -0.50f; c3 = 0.20f;  }
    if (v >  0.0f) { c0 = 0.3f;  c1 = 0.7f;  c2 =  0.10f; c3 = -0.15f; }
    if (v >  1.0f) { c0 = -0.2f; c1 = 0.4f;  c2 =  0.60f; c3 = 0.05f;  }
    if (v >  2.0f) { c0 = 1.0f;  c1 = -0.3f; c2 =  0.20f; c3 = 0.01f;  }
    float r = ((c3 * v + c2) * v + c1) * v + c0;
    return (v < -2.0f) ? 0.0f : r;
}

#if USE_TDM
typedef unsigned int u32x4 __attribute__((ext_vector_type(4)));
typedef int          i32x8 __attribute__((ext_vector_type(8)));
typedef int          i32x4 __attribute__((ext_vector_type(4)));
#endif

__global__ __launch_bounds__(256)
void pw_transpose_kernel(const float* __restrict__ x, float* __restrict__ out) {
    __shared__ float tile[TILE * LDS_STRIDE];

    const int f0 = blockIdx.x * TILE;   // column tile in x == output row tile
    const int b0 = blockIdx.y * TILE;   // row tile in x    == output col tile
    const int t  = threadIdx.x;

#if USE_TDM
    // One wave issues the TDM DMA of the 64x64 f32 tile into LDS with HW row
    // padding (65-dword row stride). TENSORcnt tracks completion.
    if ((t >> 5) == 0) {
        unsigned lds_off = (unsigned)(uintptr_t)(&tile[0]);
        unsigned long long gaddr =
            (unsigned long long)(uintptr_t)(x + (size_t)b0 * F_DIM + f0);

        u32x4 g0;
        g0.x = 1u;                                        // count=1, no gather
        g0.y = lds_off;                                   // lds_addr (bytes)
        g0.z = (unsigned)(gaddr & 0xFFFFFFFFull);         // global_addr[31:0]
        g0.w = (unsigned)((gaddr >> 32) & 0x1FFFFFFull)   // global_addr[56:32]
             | (2u << 30);                                // type = 2 (image)

        i32x8 g1;
        g1[0] = (2 << 16)      // data_size = 4 bytes
              | (1 << 20)      // pad_enable
              | (5 << 22);     // pad_interval: 64 dwords
                               // pad_amount = 0 -> 1 dword
        g1[1] = (int)((unsigned)(F_DIM & 0xFFFF) << 16);  // tensor_dim0[15:0]
        g1[2] = (int)(((unsigned)F_DIM >> 16)
              | ((unsigned)(B_DIM & 0xFFFF) << 16));      // dim0 hi | dim1 lo
        g1[3] = (int)(((unsigned)B_DIM >> 16)
              | (64u << 16));                             // dim1 hi | tile_dim0=64
        g1[4] = 64;                                       // tile_dim1=64, tile_dim2=0
        g1[5] = F_DIM;                                    // tensor_dim0_stride[31:0]
        g1[6] = 0;                                        // stride hi | dim1_stride lo
        g1[7] = 0;

        i32x4 zero4 = {0, 0, 0, 0};
        i32x8 zero8 = {0, 0, 0, 0, 0, 0, 0, 0};
        // clang-23 / therock-10.0 lane: 6-arg form
        // (g0 u32x4, g1 i32x8, g2 i32x4, g3 i32x4, extra i32x8, cpol i32)
        __builtin_amdgcn_tensor_load_to_lds(g0, g1, zero4, zero4, zero8, 0);
        __builtin_amdgcn_s_wait_tensorcnt(0);
        __asm__ volatile("" ::: "memory");
    }
    __syncthreads();
#else
    // Fallback: coalesced scalar loads along F into padded LDS rows.
    {
        const int col  = t & 63;      // f within tile
        const int rbas = t >> 6;      // 0..3
        for (int i = 0; i < TILE; i += 4) {
            int r = rbas + i;         // b within tile
            tile[r * LDS_STRIDE + col] =
                x[(size_t)(b0 + r) * F_DIM + (f0 + col)];
        }
    }
    __syncthreads();
#endif

    // Store phase: out[f, b] = piecewise(x[b, f]). float4 along b -> b128 stores.
    const int tb = (t & 15) * 4;      // b_local base (0..60)
    const int tf = t >> 4;            // f_local row  (0..15)
    for (int i = 0; i < 4; ++i) {
        int fr = tf + 16 * i;         // f within tile
        float4 r;
        r.x = piecewise(tile[(tb + 0) * LDS_STRIDE + fr]);
        r.y = piecewise(tile[(tb + 1) * LDS_STRIDE + fr]);
        r.z = piecewise(tile[(tb + 2) * LDS_STRIDE + fr]);
        r.w = piecewise(tile[(tb + 3) * LDS_STRIDE + fr]);
        *reinterpret_cast<float4*>(
            &out[(size_t)(f0 + fr) * B_DIM + (b0 + tb)]) = r;
    }
}

extern "C" void kernel_launch(void* const* d_in, const int* in_sizes, int n_in,
                              void* d_out, int out_size, void* d_ws, size_t ws_size,
                              hipStream_t stream) {
    // inputs: x [B,F] f32, w [F,F] f32 (dead code in reference), b [F] f32 (dead)
    const float* x = (const float*)d_in[0];
    float* out = (float*)d_out;          // [F, B] f32
    (void)in_sizes; (void)n_in; (void)d_ws; (void)ws_size; (void)out_size;

    dim3 grid(F_DIM / TILE, B_DIM / TILE);   // (64, 128)
    pw_transpose_kernel<<<grid, 256, 0, stream>>>(x, out);
}